// CTCLayer_20693152432755
// MI455X (gfx1250) — compile-verified
//
#include <hip/hip_runtime.h>
#include <hip/hip_bf16.h>
#include <stdint.h>
#include <stddef.h>

// Problem constants (fixed by the reference: B=256, T=256, C=1000, L=64)
#define B_N 256
#define T_N 256
#define C_N 1000
#define L_N 64
#define S_N 129          // 2L+1 extended states
#define BLANK_C 999      // blank = C-1
#define LPS 72           // padded gathered-row stride (>=65, 288B = 16B-aligned)
#define NEGINF -1e30f
#define EPSV 1e-7f

typedef __attribute__((address_space(1))) int g_int;   // global
typedef __attribute__((address_space(3))) int l_int;   // LDS

// ---------------------------------------------------------------------------
// CDNA5 async global->LDS copy (ASYNCcnt-tracked) with builtin/asm fallback.
// One float per calling lane.  gfx1250-specific path.
// ---------------------------------------------------------------------------
__device__ __forceinline__ void async_lds_load_f32(float* lds_dst, const float* gsrc) {
#if __has_builtin(__builtin_amdgcn_global_load_async_to_lds_b32)
  __builtin_amdgcn_global_load_async_to_lds_b32(
      (g_int*)(unsigned long long)gsrc,
      (l_int*)(unsigned int)(unsigned long long)lds_dst,
      0, 0);
#else
  // Flat addr low 32 bits == LDS offset on CDNA5 (aperture = high 32 bits).
  unsigned lds_off = (unsigned)(unsigned long long)lds_dst;
  asm volatile("global_load_async_to_lds_b32 %0, %1, off"
               :: "v"(lds_off), "v"(gsrc) : "memory");
#endif
}

__device__ __forceinline__ void wait_async_le(int n_is_one) {
#if __has_builtin(__builtin_amdgcn_s_wait_asynccnt)
  if (n_is_one) __builtin_amdgcn_s_wait_asynccnt(1);
  else          __builtin_amdgcn_s_wait_asynccnt(0);
#else
  if (n_is_one) asm volatile("s_wait_asynccnt 0x1" ::: "memory");
  else          asm volatile("s_wait_asynccnt 0x0" ::: "memory");
#endif
}

// numerically-stable logaddexp, matches jnp semantics for -1e30 sentinels
__device__ __forceinline__ float lse2(float a, float b) {
  float m = fmaxf(a, b);
  float d = fminf(a, b) - m;          // <= 0
  return m + log1pf(__expf(d));
}

// ---------------------------------------------------------------------------
// Kernel 1: gather + log.  One wave per (b,t) row; 65 scattered dword loads
// out of the 1000-class row, compacted to lp[(b*T+t)*LPS + j].
// This is the HBM-bandwidth-bound kernel (~230 MB read, ~17 MB write).
// ---------------------------------------------------------------------------
__global__ void ctc_gather_log(const float* __restrict__ y_pred,
                               const int*   __restrict__ y_true,
                               float*       __restrict__ lp) {
  const int row  = blockIdx.x * 8 + (threadIdx.x >> 5);   // b*T + t
  const int lane = threadIdx.x & 31;
  const int b    = row >> 8;                              // T = 256
  const float* src = y_pred + (size_t)row * C_N;
  const int*   lab = y_true + b * L_N;
  float*       dst = lp + (size_t)row * LPS;
#pragma unroll
  for (int k = 0; k < 3; ++k) {
    int j = lane + 32 * k;
    if (j < L_N + 1) {
      int idx = (j < L_N) ? lab[j] : BLANK_C;
      dst[j] = __logf(src[idx] + EPSV);
    }
  }
}

// ---------------------------------------------------------------------------
// Kernel 2: CTC forward recursion.  One block per batch, 160 threads
// (5 wave32s), one extended state per thread (tid < 129).
// lp rows double-buffered in LDS via async global->LDS loads so the next
// row streams in while the current logaddexp step computes.
// ---------------------------------------------------------------------------
__global__ void __launch_bounds__(160)
ctc_forward(const float* __restrict__ lp,
            const int*   __restrict__ y_true,
            float*       __restrict__ out) {
  const int b   = blockIdx.x;
  const int tid = threadIdx.x;

  __shared__ float lpbuf[2][LPS];
  __shared__ float alpha[2][132];

  const float* lprow = lp + (size_t)b * T_N * LPS;

  // per-state constants
  int  labIdx = L_N;      // even states -> blank slot (index 64)
  bool allow  = false;    // skip transition s-2 -> s permitted
  if (tid < S_N && (tid & 1)) {
    int li = (tid - 1) >> 1;
    labIdx = li;
    if (tid >= 3) allow = (y_true[b * L_N + li] != y_true[b * L_N + li - 1]);
  }

  // t = 0 init
  if (tid < S_N) {
    float a = NEGINF;
    if (tid == 0)      a = lprow[L_N];   // blank emission at t=0
    else if (tid == 1) a = lprow[0];     // first label at t=0
    alpha[0][tid] = a;
  }

  // prologue: stream row t=1 into lpbuf[1]
  if (tid < LPS) async_lds_load_f32(&lpbuf[1][tid], lprow + LPS + tid);
  __syncthreads();  // alpha[0] visible

  for (int t = 1; t < T_N; ++t) {
    const int cb = t & 1;          // LDS buffer holding row t
    const int ap = (t - 1) & 1;    // alpha read buffer
    const int aw = t & 1;          // alpha write buffer

    // stream row t+1 into the other buffer (freed by last iter's barrier)
    if (t + 1 < T_N) {
      if (tid < LPS)
        async_lds_load_f32(&lpbuf[cb ^ 1][tid], lprow + (size_t)(t + 1) * LPS + tid);
      wait_async_le(1);   // oldest outstanding (row t) complete; loads in order
    } else {
      wait_async_le(0);
    }
    __syncthreads();      // row t visible to all waves; alpha[ap] visible

    if (tid < S_N) {
      float a  = alpha[ap][tid];
      float s1 = (tid >= 1) ? alpha[ap][tid - 1] : NEGINF;
      float v  = lse2(a, s1);
      if (allow) v = lse2(v, alpha[ap][tid - 2]);
      alpha[aw][tid] = v + lpbuf[cb][labIdx];
    }
    __syncthreads();      // alpha[aw] published; lpbuf[cb] free for reuse
  }

  if (tid == 0) {
    // T-1 = 255 is odd -> final alphas live in buffer 1
    out[b] = -lse2(alpha[1][S_N - 1], alpha[1][S_N - 2]);
  }
}

// ---------------------------------------------------------------------------
// Fallback (only if ws too small): fused gather+recursion, no lp workspace.
// ---------------------------------------------------------------------------
__global__ void __launch_bounds__(160)
ctc_forward_fused(const float* __restrict__ y_pred,
                  const int*   __restrict__ y_true,
                  float*       __restrict__ out) {
  const int b   = blockIdx.x;
  const int tid = threadIdx.x;
  __shared__ float alpha[2][132];

  int  cls   = BLANK_C;
  bool allow = false;
  if (tid < S_N && (tid & 1)) {
    int li = (tid - 1) >> 1;
    cls = y_true[b * L_N + li];
    if (tid >= 3) allow = (cls != y_true[b * L_N + li - 1]);
  }
  const float* row0 = y_pred + (size_t)b * T_N * C_N;
  if (tid < S_N) {
    float a = NEGINF;
    if (tid <= 1) a = __logf(row0[cls] + EPSV);
    alpha[0][tid] = a;
  }
  __syncthreads();
  for (int t = 1; t < T_N; ++t) {
    const int ap = (t - 1) & 1, aw = t & 1;
    if (tid < S_N) {
      float lpv = __logf(row0[(size_t)t * C_N + cls] + EPSV);
      float a  = alpha[ap][tid];
      float s1 = (tid >= 1) ? alpha[ap][tid - 1] : NEGINF;
      float v  = lse2(a, s1);
      if (allow) v = lse2(v, alpha[ap][tid - 2]);
      alpha[aw][tid] = v + lpv;
    }
    __syncthreads();
  }
  if (tid == 0) out[b] = -lse2(alpha[1][S_N - 1], alpha[1][S_N - 2]);
}

extern "C" void kernel_launch(void* const* d_in, const int* in_sizes, int n_in,
                              void* d_out, int out_size, void* d_ws, size_t ws_size,
                              hipStream_t stream) {
  const int*   y_true = (const int*)d_in[0];    // [B, L] int
  const float* y_pred = (const float*)d_in[1];  // [B, T, C] f32 softmax probs
  float* out = (float*)d_out;                   // [B, 1] f32

  const size_t lp_bytes = (size_t)B_N * T_N * LPS * sizeof(float);  // 18.9 MB
  if (ws_size >= lp_bytes) {
    float* lp = (float*)d_ws;
    ctc_gather_log<<<dim3((B_N * T_N) / 8), dim3(256), 0, stream>>>(y_pred, y_true, lp);
    ctc_forward<<<dim3(B_N), dim3(160), 0, stream>>>(lp, y_true, out);
  } else {
    ctc_forward_fused<<<dim3(B_N), dim3(160), 0, stream>>>(y_pred, y_true, out);
  }
}